// MultiHeadAttention_62732292325710
// MI455X (gfx1250) — compile-verified
//
#include <hip/hip_runtime.h>

typedef __attribute__((ext_vector_type(2))) float v2f;
typedef __attribute__((ext_vector_type(8))) float v8f;

#define B_ 4
#define S_ 2048
#define E_ 1024
#define H_ 16
#define D_ 64

__device__ __forceinline__ v8f wmma_f32(v2f a, v2f b, v8f c) {
  // V_WMMA_F32_16X16X4_F32: D = A(16x4,f32) x B(4x16,f32) + C(16x16,f32)
  return __builtin_amdgcn_wmma_f32_16x16x4_f32(
      /*neg_a=*/false, a, /*neg_b=*/false, b,
      /*c_mod=*/(short)0, c, /*reuse_a=*/false, /*reuse_b=*/false);
}

// ---------------------------------------------------------------------------
// Kernel 1: fused QKV projection.  GEMM: [8192 x 1024] x [1024 x (H*D)]
// grid = (16 heads, 64 row-tiles, 3 {Q,K,V}), block = 256 (8 waves)
// Block tile 128x64 (BN = 64 = D, so one head per block), BK = 32.
// Each wave computes a 32x32 patch (2x2 WMMA accumulators).
// Output layout [B,H,S,D] for the attention kernel.
// ---------------------------------------------------------------------------
__global__ __launch_bounds__(256) void qkv_proj_kernel(
    const float* __restrict__ x,
    const float* __restrict__ Wq, const float* __restrict__ bq,
    const float* __restrict__ Wk, const float* __restrict__ bk,
    const float* __restrict__ Wv, const float* __restrict__ bv,
    float* __restrict__ Qb, float* __restrict__ Kb, float* __restrict__ Vb) {
  const int h  = blockIdx.x;        // head -> N tile of 64
  const int m0 = blockIdx.y * 128;  // row tile
  const int z  = blockIdx.z;        // 0=Q 1=K 2=V
  const float* W    = (z == 0) ? Wq : (z == 1) ? Wk : Wv;
  const float* bias = (z == 0) ? bq : (z == 1) ? bk : bv;
  float*       Out  = (z == 0) ? Qb : (z == 1) ? Kb : Vb;

  __shared__ float As[128][36];  // [m][e]
  __shared__ float Bs[64][36];   // [d][e]  (W tile transposed)

  const int tid  = threadIdx.x;
  const int wave = tid >> 5;
  const int lane = tid & 31;
  const int l16  = lane & 15;
  const int lh   = lane >> 4;
  const int wm   = wave >> 1;  // 0..3 -> rows 32*wm
  const int wn   = wave & 1;   // 0..1 -> cols 32*wn

  v8f acc[2][2] = {};

  for (int k0 = 0; k0 < E_; k0 += 32) {
    // stage A: x[m0..m0+127][k0..k0+31]
    {
      const int col4 = (tid & 7) * 4;
      const int mb   = tid >> 3;  // 0..31
#pragma unroll
      for (int r = 0; r < 4; ++r) {
        const int m = mb + r * 32;
        *(float4*)&As[m][col4] =
            *(const float4*)&x[(size_t)(m0 + m) * E_ + k0 + col4];
      }
    }
    // stage B (transposed): Bs[d][e] = W[h][k0+e][d]
    {
      const int d  = tid & 63;
      const int e0 = (tid >> 6) * 8;
#pragma unroll
      for (int r = 0; r < 8; ++r) {
        const int e = e0 + r;
        Bs[d][e] = W[((size_t)h * E_ + (k0 + e)) * D_ + d];
      }
    }
    __syncthreads();

#pragma unroll
    for (int kk = 0; kk < 32; kk += 4) {
      v2f a[2], b[2];
#pragma unroll
      for (int tm = 0; tm < 2; ++tm)
        a[tm] = *(const v2f*)&As[32 * wm + 16 * tm + l16][kk + 2 * lh];
#pragma unroll
      for (int tn = 0; tn < 2; ++tn)
        b[tn] = *(const v2f*)&Bs[32 * wn + 16 * tn + l16][kk + 2 * lh];
#pragma unroll
      for (int tm = 0; tm < 2; ++tm)
#pragma unroll
        for (int tn = 0; tn < 2; ++tn)
          acc[tm][tn] = wmma_f32(a[tm], b[tn], acc[tm][tn]);
    }
    __syncthreads();
  }

  // epilogue: add bias, scatter to [B,H,S,D]
#pragma unroll
  for (int tm = 0; tm < 2; ++tm)
#pragma unroll
    for (int tn = 0; tn < 2; ++tn)
#pragma unroll
      for (int v = 0; v < 8; ++v) {
        const int m = m0 + 32 * wm + 16 * tm + v + 8 * lh;
        const int d = 32 * wn + 16 * tn + l16;
        const int bb = m >> 11;       // m / S_
        const int ss = m & (S_ - 1);  // m % S_
        Out[(((size_t)bb * H_ + h) * S_ + ss) * D_ + d] =
            acc[tm][tn][v] + bias[h * D_ + d];
      }
}

// ---------------------------------------------------------------------------
// Kernel 2: flash attention, causal.  grid = (32 q-blocks, 64 b*h),
// block = 128 (4 waves).  Each wave owns 16 query rows x all 64 dims.
// Q fragments (pre-scaled by 1/sqrt(D)) live in registers; K/V tiles in LDS.
// Online softmax state per row (max, sum) replicated across 16-lane groups.
// Writes concat layout [B, S, H*D].
// ---------------------------------------------------------------------------
__global__ __launch_bounds__(128) void attn_kernel(
    const float* __restrict__ Qb, const float* __restrict__ Kb,
    const float* __restrict__ Vb, float* __restrict__ conc) {
  const int qblk = blockIdx.x;
  const int bh   = blockIdx.y;
  const int b    = bh >> 4;
  const int h    = bh & 15;
  const int row0 = qblk * 64;
  const size_t base = (size_t)bh * S_ * D_;

  __shared__ float Ks[64][68];
  __shared__ float Vs[64][68];
  __shared__ float Ps[4][16][68];  // per-wave P pane

  const int tid  = threadIdx.x;
  const int wave = tid >> 5;  // query rows 16*wave .. +15
  const int lane = tid & 31;
  const int l16  = lane & 15;
  const int lh   = lane >> 4;

  // stage Q through Ks, then keep scaled fragments in registers
  {
    const int d4 = (tid & 15) * 4;
    const int rb = tid >> 4;  // 0..7
#pragma unroll
    for (int it = 0; it < 8; ++it) {
      const int r = rb + it * 8;
      *(float4*)&Ks[r][d4] =
          *(const float4*)&Qb[base + (size_t)(row0 + r) * D_ + d4];
    }
  }
  __syncthreads();
  v2f qf[16];
#pragma unroll
  for (int kk = 0; kk < 64; kk += 4) {
    v2f q = *(const v2f*)&Ks[16 * wave + l16][kk + 2 * lh];
    qf[kk >> 2] = q * 0.125f;  // 1/sqrt(64)
  }
  __syncthreads();

  const float NEG_INF = -__builtin_inff();
  v8f o[4] = {};
  float mrow[8], lrow[8];
#pragma unroll
  for (int v = 0; v < 8; ++v) { mrow[v] = NEG_INF; lrow[v] = 0.0f; }

  for (int j = 0; j <= qblk; ++j) {
    {  // stage K, V block j
      const int d4 = (tid & 15) * 4;
      const int rb = tid >> 4;
#pragma unroll
      for (int it = 0; it < 8; ++it) {
        const int r = rb + it * 8;
        const size_t g = base + (size_t)(j * 64 + r) * D_ + d4;
        *(float4*)&Ks[r][d4] = *(const float4*)&Kb[g];
        *(float4*)&Vs[r][d4] = *(const float4*)&Vb[g];
      }
    }
    __syncthreads();

    // S = Q * K^T  (scaled)
    v8f s[4] = {};
#pragma unroll
    for (int kk = 0; kk < 64; kk += 4) {
      const v2f a = qf[kk >> 2];
#pragma unroll
      for (int tn = 0; tn < 4; ++tn) {
        const v2f bb = *(const v2f*)&Ks[16 * tn + l16][kk + 2 * lh];
        s[tn] = wmma_f32(a, bb, s[tn]);
      }
    }

    if (j == qblk) {  // causal mask on the diagonal block
#pragma unroll
      for (int tn = 0; tn < 4; ++tn)
#pragma unroll
        for (int v = 0; v < 8; ++v) {
          const int r = 16 * wave + v + 8 * lh;
          const int c = 16 * tn + l16;
          if (c > r) s[tn][v] = NEG_INF;
        }
    }

    // online softmax
    float rmax[8];
#pragma unroll
    for (int v = 0; v < 8; ++v)
      rmax[v] = fmaxf(fmaxf(s[0][v], s[1][v]), fmaxf(s[2][v], s[3][v]));
#pragma unroll
    for (int off = 1; off < 16; off <<= 1)
#pragma unroll
      for (int v = 0; v < 8; ++v)
        rmax[v] = fmaxf(rmax[v], __shfl_xor(rmax[v], off, 32));

    float alpha[8];
#pragma unroll
    for (int v = 0; v < 8; ++v) {
      const float mn = fmaxf(mrow[v], rmax[v]);
      alpha[v] = __expf(mrow[v] - mn);
      mrow[v]  = mn;
    }
#pragma unroll
    for (int tn = 0; tn < 4; ++tn)
#pragma unroll
      for (int v = 0; v < 8; ++v) s[tn][v] = __expf(s[tn][v] - mrow[v]);

    float rsum[8];
#pragma unroll
    for (int v = 0; v < 8; ++v)
      rsum[v] = (s[0][v] + s[1][v]) + (s[2][v] + s[3][v]);
#pragma unroll
    for (int off = 1; off < 16; off <<= 1)
#pragma unroll
      for (int v = 0; v < 8; ++v) rsum[v] += __shfl_xor(rsum[v], off, 32);
#pragma unroll
    for (int v = 0; v < 8; ++v) lrow[v] = lrow[v] * alpha[v] + rsum[v];

#pragma unroll
    for (int tn = 0; tn < 4; ++tn)
#pragma unroll
      for (int v = 0; v < 8; ++v) o[tn][v] *= alpha[v];

    // P -> per-wave LDS pane (C layout -> row major)
#pragma unroll
    for (int tn = 0; tn < 4; ++tn)
#pragma unroll
      for (int v = 0; v < 8; ++v)
        Ps[wave][v + 8 * lh][16 * tn + l16] = s[tn][v];

    // O += P * V
#pragma unroll
    for (int kk = 0; kk < 64; kk += 4) {
      const v2f a = *(const v2f*)&Ps[wave][l16][kk + 2 * lh];
#pragma unroll
      for (int tn = 0; tn < 4; ++tn) {
        v2f bb;
        bb.x = Vs[kk + 2 * lh + 0][16 * tn + l16];
        bb.y = Vs[kk + 2 * lh + 1][16 * tn + l16];
        o[tn] = wmma_f32(a, bb, o[tn]);
      }
    }
    __syncthreads();  // before next iteration overwrites Ks/Vs
  }

  // normalize, write concat layout [B, S, H*D]
  float rl[8];
#pragma unroll
  for (int v = 0; v < 8; ++v) rl[v] = 1.0f / lrow[v];
#pragma unroll
  for (int tn = 0; tn < 4; ++tn)
#pragma unroll
    for (int v = 0; v < 8; ++v) {
      const int r = row0 + 16 * wave + v + 8 * lh;
      const int c = 16 * tn + l16;
      conc[((size_t)b * S_ + r) * E_ + h * D_ + c] = o[tn][v] * rl[v];
    }
}

// ---------------------------------------------------------------------------
// Kernel 3: output projection.  out = conc @ Wo^T + bo
// Same tiling as kernel 1; Wo rows are contiguous in k, so Bs[n][k] loads
// are coalesced without a transpose.
// ---------------------------------------------------------------------------
__global__ __launch_bounds__(256) void out_proj_kernel(
    const float* __restrict__ conc, const float* __restrict__ Wo,
    const float* __restrict__ bo, float* __restrict__ out) {
  const int n0 = blockIdx.x * 64;
  const int m0 = blockIdx.y * 128;

  __shared__ float As[128][36];
  __shared__ float Bs[64][36];  // [n][k]

  const int tid  = threadIdx.x;
  const int wave = tid >> 5;
  const int lane = tid & 31;
  const int l16  = lane & 15;
  const int lh   = lane >> 4;
  const int wm   = wave >> 1;
  const int wn   = wave & 1;

  v8f acc[2][2] = {};

  for (int k0 = 0; k0 < E_; k0 += 32) {
    {
      const int col4 = (tid & 7) * 4;
      const int mb   = tid >> 3;
#pragma unroll
      for (int r = 0; r < 4; ++r) {
        const int m = mb + r * 32;
        *(float4*)&As[m][col4] =
            *(const float4*)&conc[(size_t)(m0 + m) * E_ + k0 + col4];
      }
    }
    {
      const int n  = tid >> 2;        // 0..63
      const int kc = (tid & 3) * 8;   // 0,8,16,24
      *(float4*)&Bs[n][kc] =
          *(const float4*)&Wo[(size_t)(n0 + n) * E_ + k0 + kc];
      *(float4*)&Bs[n][kc + 4] =
          *(const float4*)&Wo[(size_t)(n0 + n) * E_ + k0 + kc + 4];
    }
    __syncthreads();

#pragma unroll
    for (int kk = 0; kk < 32; kk += 4) {
      v2f a[2], b[2];
#pragma unroll
      for (int tm = 0; tm < 2; ++tm)
        a[tm] = *(const v2f*)&As[32 * wm + 16 * tm + l16][kk + 2 * lh];
#pragma unroll
      for (int tn = 0; tn < 2; ++tn)
        b[tn] = *(const v2f*)&Bs[32 * wn + 16 * tn + l16][kk + 2 * lh];
#pragma unroll
      for (int tm = 0; tm < 2; ++tm)
#pragma unroll
        for (int tn = 0; tn < 2; ++tn)
          acc[tm][tn] = wmma_f32(a[tm], b[tn], acc[tm][tn]);
    }
    __syncthreads();
  }

#pragma unroll
  for (int tm = 0; tm < 2; ++tm)
#pragma unroll
    for (int tn = 0; tn < 2; ++tn)
#pragma unroll
      for (int v = 0; v < 8; ++v) {
        const int m = m0 + 32 * wm + 16 * tm + v + 8 * lh;
        const int c = n0 + 32 * wn + 16 * tn + l16;
        out[(size_t)m * E_ + c] = acc[tm][tn][v] + bo[c];
      }
}

// ---------------------------------------------------------------------------
extern "C" void kernel_launch(void* const* d_in, const int* in_sizes, int n_in,
                              void* d_out, int out_size, void* d_ws,
                              size_t ws_size, hipStream_t stream) {
  const float* x  = (const float*)d_in[0];
  const float* Wq = (const float*)d_in[1];
  const float* bq = (const float*)d_in[2];
  const float* Wk = (const float*)d_in[3];
  const float* bk = (const float*)d_in[4];
  const float* Wv = (const float*)d_in[5];
  const float* bv = (const float*)d_in[6];
  const float* Wo = (const float*)d_in[7];
  const float* bo = (const float*)d_in[8];
  float* out = (float*)d_out;

  float* ws = (float*)d_ws;
  const size_t sz = (size_t)B_ * H_ * S_ * D_;  // 8M floats = 32 MB
  float* Qb   = ws;
  float* Kb   = ws + sz;
  float* Vb   = ws + 2 * sz;
  float* conc = ws + 3 * sz;

  qkv_proj_kernel<<<dim3(H_, (B_ * S_) / 128, 3), 256, 0, stream>>>(
      x, Wq, bq, Wk, bk, Wv, bv, Qb, Kb, Vb);
  attn_kernel<<<dim3(S_ / 64, B_ * H_), 128, 0, stream>>>(Qb, Kb, Vb, conc);
  out_proj_kernel<<<dim3(E_ / 64, (B_ * S_) / 128), 256, 0, stream>>>(
      conc, Wo, bo, out);
}